// AttentionLinearBiases_67757404061806
// MI455X (gfx1250) — compile-verified
//
#include <hip/hip_runtime.h>

// out[b,h,q,k] = in[b,h,q,k] + slope[h] * (k - (S-1)),  S = 4096, H = 16.
// Pure HBM-streaming kernel: 1 GiB in + 1 GiB out, 1 FMA/element.
// Strategy: b128 non-temporal loads/stores, wave-uniform slope, one block per
// (head, query) row so the bias term is an affine function of threadIdx.x.

typedef float f4 __attribute__((ext_vector_type(4)));

// slopes[h] = 2^(-8*(h+1)/16) = 2^(-0.5*(h+1)), computed in f64 and rounded
// to f32 (matches numpy double -> float32 cast in the reference).
__constant__ float kSlopes[16] = {
    0.70710678118654752f, 0.5f,
    0.35355339059327376f, 0.25f,
    0.17677669529663688f, 0.125f,
    0.08838834764831844f, 0.0625f,
    0.04419417382415922f, 0.03125f,
    0.02209708691207961f, 0.015625f,
    0.01104854345603981f, 0.0078125f,
    0.00552427172801990f, 0.00390625f};

__global__ __launch_bounds__(256) void alibi_add_kernel(
    const float* __restrict__ in, float* __restrict__ out) {
  const int t = threadIdx.x;
  const unsigned row = blockIdx.x;     // row = h*4096 + q, 65536 rows total
  const unsigned h = row >> 12;        // 4096 query rows per head
  const float slope = kSlopes[h];      // wave-uniform -> SGPR

  const size_t base4 = (size_t)row * 1024u;  // float4 index of row start
  const f4* __restrict__ in4 = (const f4*)in + base4;
  f4* __restrict__ out4 = (f4*)out + base4;

  // Each thread: 4 x float4 = 16 floats; block covers the full 4096-key row.
  // k for element j,lane t is 4*(t + 256*j) + {0,1,2,3}; distance = k - 4095.
  f4 v[4];
#pragma unroll
  for (int j = 0; j < 4; ++j) {
    // Non-temporal: stream-once data, keep it out of L2 retention (TH=NT).
    v[j] = __builtin_nontemporal_load(in4 + (t + j * 256));
  }
#pragma unroll
  for (int j = 0; j < 4; ++j) {
    const float d0 = (float)(4 * (t + j * 256) - 4095);
    f4 r;
    r.x = fmaf(slope, d0 + 0.0f, v[j].x);
    r.y = fmaf(slope, d0 + 1.0f, v[j].y);
    r.z = fmaf(slope, d0 + 2.0f, v[j].z);
    r.w = fmaf(slope, d0 + 3.0f, v[j].w);
    __builtin_nontemporal_store(r, out4 + (t + j * 256));
  }
}

extern "C" void kernel_launch(void* const* d_in, const int* in_sizes, int n_in,
                              void* d_out, int out_size, void* d_ws, size_t ws_size,
                              hipStream_t stream) {
  (void)in_sizes; (void)n_in; (void)d_ws; (void)ws_size; (void)out_size;
  const float* scores = (const float*)d_in[0];
  float* out = (float*)d_out;
  // 16 heads * 4096 query rows = 65536 blocks, each handling one 4096-key row.
  dim3 grid(65536), block(256);
  alibi_add_kernel<<<grid, block, 0, stream>>>(scores, out);
}